// UnionRGCNLayer_74431783240012
// MI455X (gfx1250) — compile-verified
//
#include <hip/hip_runtime.h>
#include <math.h>

// ---------------------------------------------------------------------------
// RGCN layer for MI455X (gfx1250, wave32, WMMA)
//   hw   = h @ Wn                      (WMMA f32 GEMM)
//   rw   = rel_emb @ Wn                (WMMA f32 GEMM)
//   loopm= select(norm>0, h@Wl, h@We)  (dual-B WMMA GEMM, fused select)
//   skip = sigmoid(h@Ws + bias)        (WMMA GEMM, fused epilogue)
//   agg  = scatter-add over edges of hw[src]+rw[etype]   (f32 HW atomics, L2)
//   out  = skip*(LN(agg*norm)*g+b + loopm) + (1-skip)*h  (one wave per node)
// ---------------------------------------------------------------------------

typedef __attribute__((ext_vector_type(2))) float v2f;
typedef __attribute__((ext_vector_type(8))) float v8f;
typedef __attribute__((ext_vector_type(4))) int   v4i;

#define LDS_STRIDE 132  // 128 + 4 pad: removes 16-way bank conflicts on col reads

#if defined(__has_builtin)
#if __has_builtin(__builtin_amdgcn_global_load_async_to_lds_b128)
#define HAVE_ASYNC_LDS 1
#endif
#endif

__device__ __forceinline__ void wait_async0()
{
#if defined(__has_builtin)
#if __has_builtin(__builtin_amdgcn_s_wait_asynccnt)
    __builtin_amdgcn_s_wait_asynccnt(0);
    return;
#else
    asm volatile("s_wait_asynccnt 0" ::: "memory");
    return;
#endif
#else
    asm volatile("s_wait_asynccnt 0" ::: "memory");
#endif
}

// MODE 0: C = A@B0
// MODE 1: C = (norm[row]>0) ? A@B0 : A@B1
// MODE 2: C = sigmoid(A@B0 + bias[col])
// EXACT : M is a multiple of 16 -> no store guard needed
template <int MODE, bool EXACT>
__global__ __launch_bounds__(256) void gemm_rgcn(
    const float* __restrict__ A,      // [M,128]
    const float* __restrict__ B0,     // [128,128]
    const float* __restrict__ B1,     // [128,128] (MODE 1)
    const float* __restrict__ normv,  // [M]       (MODE 1)
    const float* __restrict__ bias,   // [128]     (MODE 2)
    float* __restrict__ C,            // [M,128]
    int M)
{
    __shared__ float As[16 * LDS_STRIDE];

    const int m0  = blockIdx.x * 16;
    const int tid = threadIdx.x;

    // Stage the 16x128 A tile into LDS (coalesced, row-clamped).
    {
        int row  = tid >> 4;          // 0..15
        int col8 = (tid & 15) * 8;    // 0..120
        int gr   = m0 + row;
        if (gr > M - 1) gr = M - 1;
        const float* gp = A + (size_t)gr * 128 + col8;
        float*       lp = &As[row * LDS_STRIDE + col8];
#ifdef HAVE_ASYNC_LDS
        // CDNA5 async DMA: global -> LDS without VGPR round-trip (ASYNCcnt).
        __builtin_amdgcn_global_load_async_to_lds_b128((v4i*)gp, (v4i*)lp, 0, 0);
        __builtin_amdgcn_global_load_async_to_lds_b128((v4i*)gp, (v4i*)lp, 16, 0);
        wait_async0();
#else
        const float4* srcp = (const float4*)gp;
        float4 v0 = srcp[0];
        float4 v1 = srcp[1];
        float4* dstp = (float4*)lp;
        dstp[0] = v0;
        dstp[1] = v1;
#endif
    }
    __syncthreads();

    const int lane    = tid & 31;
    const int n       = lane & 15;   // N index within 16x16 tile
    const int half    = lane >> 4;   // selects K pair for A/B frags
    const int colbase = (tid >> 5) * 16;  // 8 waves cover all 128 columns

    v8f acc0 = {};
    v8f acc1 = {};

#pragma unroll 8
    for (int k = 0; k < 128; k += 4) {
        const int kk = k + 2 * half;
        // A frag: lane%16 = M row, VGPR0/1 = consecutive K (ISA 16x4 layout)
        v2f a;
        a.x = As[n * LDS_STRIDE + kk];
        a.y = As[n * LDS_STRIDE + kk + 1];
        // B frag: lane%16 = N col, lane half selects K pair
        v2f b;
        b.x = B0[(size_t)kk * 128 + colbase + n];
        b.y = B0[(size_t)(kk + 1) * 128 + colbase + n];
        acc0 = __builtin_amdgcn_wmma_f32_16x16x4_f32(
            false, a, false, b, (short)0, acc0, false, false);
        if (MODE == 1) {
            v2f b2;
            b2.x = B1[(size_t)kk * 128 + colbase + n];
            b2.y = B1[(size_t)(kk + 1) * 128 + colbase + n];
            acc1 = __builtin_amdgcn_wmma_f32_16x16x4_f32(
                false, a, false, b2, (short)0, acc1, false, false);
        }
    }

    // Epilogue: C/D layout -> VGPR j holds rows (j, j+8) split by lane half.
    float* cp = C + (size_t)(m0 + 8 * half) * 128 + colbase + n;
#pragma unroll
    for (int j = 0; j < 8; ++j) {
        const int row = m0 + j + 8 * half;
        if (EXACT || row < M) {
            float v = acc0[j];
            if (MODE == 1) {
                v = (normv[row] > 0.0f) ? v : acc1[j];
            } else if (MODE == 2) {
                float s = v + bias[colbase + n];
                v = 1.0f / (1.0f + __expf(-s));
            }
            cp[(size_t)j * 128] = v;
        }
    }
}

__global__ __launch_bounds__(256) void zero_f4(float4* __restrict__ p, int n4)
{
    int i = blockIdx.x * 256 + threadIdx.x;
    if (i < n4) p[i] = float4{0.f, 0.f, 0.f, 0.f};
}

// One wave32 per edge: lane owns a float4 chunk (32 lanes * 4 = 128 feats).
// hw/rw/agg are L2-resident on MI455X (192MB L2) -> gathers+atomics stay on-chip.
__global__ __launch_bounds__(256) void edge_scatter(
    const float4* __restrict__ hw4,   // [N,32] float4 view
    const float4* __restrict__ rw4,   // [R,32]
    const int* __restrict__ src,
    const int* __restrict__ dst,
    const int* __restrict__ etype,
    float* __restrict__ agg,          // [N,128]
    int E)
{
    const int e = blockIdx.x * 8 + (threadIdx.x >> 5);
    if (e >= E) return;
    const int lane = threadIdx.x & 31;
    const int s = src[e];
    const int d = dst[e];
    const int t = etype[e];
    float4 a = hw4[(size_t)s * 32 + lane];
    float4 b = rw4[(size_t)t * 32 + lane];
    float* p = agg + (size_t)d * 128 + lane * 4;
    unsafeAtomicAdd(p + 0, a.x + b.x);
    unsafeAtomicAdd(p + 1, a.y + b.y);
    unsafeAtomicAdd(p + 2, a.z + b.z);
    unsafeAtomicAdd(p + 3, a.w + b.w);
}

// One wave32 per node: LayerNorm + skip gate + self-loop combine.
__global__ __launch_bounds__(256) void node_final(
    const float* __restrict__ agg,
    const float* __restrict__ normv,
    const float* __restrict__ loopm,
    const float* __restrict__ skipw,
    const float* __restrict__ h,
    const float* __restrict__ gamma,
    const float* __restrict__ beta,
    float* __restrict__ out,
    int N)
{
    const int i = blockIdx.x * 8 + (threadIdx.x >> 5);
    if (i >= N) return;
    const int lane = threadIdx.x & 31;
    const size_t base = (size_t)i * 128 + lane * 4;

    float4 a = *(const float4*)(agg + base);
    const float nv = normv[i];
    float4 hn = {a.x * nv, a.y * nv, a.z * nv, a.w * nv};

    float s = hn.x + hn.y + hn.z + hn.w;
#pragma unroll
    for (int off = 16; off > 0; off >>= 1) s += __shfl_xor(s, off, 32);
    const float mu = s * 0.0078125f;  // /128

    float4 d = {hn.x - mu, hn.y - mu, hn.z - mu, hn.w - mu};
    float v = d.x * d.x + d.y * d.y + d.z * d.z + d.w * d.w;
#pragma unroll
    for (int off = 16; off > 0; off >>= 1) v += __shfl_xor(v, off, 32);
    const float rs = rsqrtf(v * 0.0078125f + 1e-5f);

    float4 g  = *(const float4*)(gamma + lane * 4);
    float4 bt = *(const float4*)(beta + lane * 4);
    float4 lm = *(const float4*)(loopm + base);
    float4 sw = *(const float4*)(skipw + base);
    float4 hv = *(const float4*)(h + base);

    float4 o;
    o.x = sw.x * ((d.x * rs * g.x + bt.x) + lm.x) + (1.f - sw.x) * hv.x;
    o.y = sw.y * ((d.y * rs * g.y + bt.y) + lm.y) + (1.f - sw.y) * hv.y;
    o.z = sw.z * ((d.z * rs * g.z + bt.z) + lm.z) + (1.f - sw.z) * hv.z;
    o.w = sw.w * ((d.w * rs * g.w + bt.w) + lm.w) + (1.f - sw.w) * hv.w;

    *(float4*)(out + base) = o;
}

extern "C" void kernel_launch(void* const* d_in, const int* in_sizes, int n_in,
                              void* d_out, int out_size, void* d_ws, size_t ws_size,
                              hipStream_t stream)
{
    const float* h     = (const float*)d_in[0];
    const float* normv = (const float*)d_in[1];
    const float* rel   = (const float*)d_in[2];
    const float* Wn    = (const float*)d_in[3];
    const float* Wl    = (const float*)d_in[4];
    const float* We    = (const float*)d_in[5];
    const float* Ws    = (const float*)d_in[6];
    const float* bias  = (const float*)d_in[7];
    const float* gamma = (const float*)d_in[8];
    const float* beta  = (const float*)d_in[9];
    const int*   src   = (const int*)d_in[10];
    const int*   dst   = (const int*)d_in[11];
    const int*   etype = (const int*)d_in[12];

    const int N = in_sizes[0] / 128;
    const int R = in_sizes[2] / 128;
    const int E = in_sizes[10];

    const size_t nd = (size_t)N * 128;
    float* ws    = (float*)d_ws;
    float* hw    = ws;            // [N,128]
    float* loopm = ws + nd;       // [N,128]
    float* skipw = ws + 2 * nd;   // [N,128]
    float* agg   = ws + 3 * nd;   // [N,128]
    float* rw    = ws + 4 * nd;   // [R,128]

    float* out = (float*)d_out;

    const dim3 blk(256);
    const bool n_exact = (N % 16) == 0;

    // Dense projections (WMMA f32)
    if (n_exact) {
        gemm_rgcn<0, true><<<dim3(N / 16), blk, 0, stream>>>(h, Wn, nullptr, nullptr, nullptr, hw, N);
        gemm_rgcn<1, true><<<dim3(N / 16), blk, 0, stream>>>(h, Wl, We, normv, nullptr, loopm, N);
        gemm_rgcn<2, true><<<dim3(N / 16), blk, 0, stream>>>(h, Ws, nullptr, nullptr, bias, skipw, N);
    } else {
        gemm_rgcn<0, false><<<dim3((N + 15) / 16), blk, 0, stream>>>(h, Wn, nullptr, nullptr, nullptr, hw, N);
        gemm_rgcn<1, false><<<dim3((N + 15) / 16), blk, 0, stream>>>(h, Wl, We, normv, nullptr, loopm, N);
        gemm_rgcn<2, false><<<dim3((N + 15) / 16), blk, 0, stream>>>(h, Ws, nullptr, nullptr, bias, skipw, N);
    }
    gemm_rgcn<0, false><<<dim3((R + 15) / 16), blk, 0, stream>>>(rel, Wn, nullptr, nullptr, nullptr, rw, R);

    // Clear accumulator, scatter messages, finalize nodes.
    const int n4 = (int)(nd / 4);
    zero_f4<<<dim3((n4 + 255) / 256), blk, 0, stream>>>((float4*)agg, n4);
    edge_scatter<<<dim3((E + 7) / 8), blk, 0, stream>>>(
        (const float4*)hw, (const float4*)rw, src, dst, etype, agg, E);
    node_final<<<dim3((N + 7) / 8), blk, 0, stream>>>(
        agg, normv, loopm, skipw, h, gamma, beta, out, N);
}